// HardwareEfficientU_71459665871622
// MI455X (gfx1250) — compile-verified
//
#include <hip/hip_runtime.h>

#define NQ 20
#define DIM (1u << NQ)

typedef __attribute__((ext_vector_type(2))) float v2f;
typedef __attribute__((ext_vector_type(8))) float v8f;

// LDS pad: +1 float per 16, +1 per 256 -> strides 1/16/256 become 1/17/273 (conflict-free-ish)
__device__ __forceinline__ int pd(int i) { return i + (i >> 4) + (i >> 8); }

// ---------------------------------------------------------------------------
// Build the 20 fused 16x16 complex group matrices:
//   layer l, group g covers index bits p=4g..4g+3  (bit p <-> big-endian qubit 19-p)
//   per-qubit gate: G = RX(tx) * RZ(tz) * (H if l==0)
//   M[i][j] = prod_b G_b[bit_b(i), bit_b(j)]
// Layout: mats[(l*5+g)*512 + {0:Re,256:Im} + i*16 + j]
// ---------------------------------------------------------------------------
__global__ void build_mats(const float* __restrict__ tx, const float* __restrict__ tz,
                           float* __restrict__ mats)
{
    int lg = blockIdx.x;            // 0..19
    int l = lg / 5, g = lg % 5;
    int t = threadIdx.x;            // 0..255
    int i = t >> 4, j = t & 15;
    float ar = 1.f, ai = 0.f;
    const float is2 = 0.70710678118654752440f;
    for (int b = 0; b < 4; ++b) {
        int p = 4 * g + b;
        int q = (NQ - 1) - p;
        float a = tx[l * NQ + q];
        float z = tz[l * NQ + q];
        float c  = cosf(a * 0.5f), s  = sinf(a * 0.5f);
        float cz = cosf(z * 0.5f), sz = sinf(z * 0.5f);
        // RX*RZ
        float g00r =  c * cz, g00i = -c * sz;
        float g01r =  s * sz, g01i = -s * cz;
        float g10r = -s * sz, g10i = -s * cz;
        float g11r =  c * cz, g11i =  c * sz;
        if (l == 0) { // right-multiply by H (H applied to state first)
            float n00r = (g00r + g01r) * is2, n00i = (g00i + g01i) * is2;
            float n01r = (g00r - g01r) * is2, n01i = (g00i - g01i) * is2;
            float n10r = (g10r + g11r) * is2, n10i = (g10i + g11i) * is2;
            float n11r = (g10r - g11r) * is2, n11i = (g10i - g11i) * is2;
            g00r = n00r; g00i = n00i; g01r = n01r; g01i = n01i;
            g10r = n10r; g10i = n10i; g11r = n11r; g11i = n11i;
        }
        int bi = (i >> b) & 1, bj = (j >> b) & 1;
        float er = bi ? (bj ? g11r : g10r) : (bj ? g01r : g00r);
        float ei = bi ? (bj ? g11i : g10i) : (bj ? g01i : g00i);
        float nar = ar * er - ai * ei;
        float nai = ar * ei + ai * er;
        ar = nar; ai = nai;
    }
    mats[(size_t)lg * 512 +       i * 16 + j] = ar;
    mats[(size_t)lg * 512 + 256 + i * 16 + j] = ai;
}

// ---------------------------------------------------------------------------
// State pass. 4096 complex amplitudes per block in LDS (split re/im, padded).
//  passB==0 : contiguous 2^12 chunk; apply groups with LDS strides 1,16,256
//             (index bits 0..11). Store contiguous.
//  passB==1 : chunk = {all 256 values of bits 12..19} x {16 consecutive low};
//             LDS index a = h*16 + t, so orig bits 12..19 live at LDS bits 4..11
//             -> apply groups with LDS strides 16,256. Store with the fused
//             CNOT-chain permutation: out[grayDecode(i)] = v[i].
// Each 16x16 complex group matrix is applied via V_WMMA_F32_16X16X4_F32:
//  per 16-vector tile, K=16 accumulated as 4 chunks; complex via 4 real WMMAs
//  (Re: Mr*Xr + (-Mi)*Xi ; Im: Mr*Xi + Mi*Xr).
// ---------------------------------------------------------------------------
__global__ __launch_bounds__(256)
void qpass(const float* __restrict__ srcRe, const float* __restrict__ srcIm,
           const float2* __restrict__ srcC, float2* __restrict__ dst,
           const float* __restrict__ mats, int matBase, int passB, int planes)
{
    __shared__ float LR[4368];
    __shared__ float LI[4368];
    const int tid   = threadIdx.x;
    const int lane  = tid & 31;
    const int wv    = tid >> 5;
    const int state = blockIdx.x >> 8;
    const int chunk = blockIdx.x & 255;
    const size_t sbase = (size_t)state << NQ;

    // ---- load global -> LDS ----
    if (!passB) {
        const size_t gb = sbase + ((size_t)chunk << 12);
        if (planes) {
            for (int r = 0; r < 16; ++r) {
                int a = r * 256 + tid;
                LR[pd(a)] = srcRe[gb + a];
                LI[pd(a)] = srcIm[gb + a];
            }
        } else {
            for (int r = 0; r < 16; ++r) {
                int a = r * 256 + tid;
                float2 v = srcC[gb + a];
                LR[pd(a)] = v.x; LI[pd(a)] = v.y;
            }
        }
    } else {
        const int lowBase = chunk << 4;
        for (int r = 0; r < 16; ++r) {
            int a = r * 256 + tid;
            int h = a >> 4, tt = a & 15;
            float2 v = srcC[sbase + ((size_t)h << 12) + lowBase + tt];
            LR[pd(a)] = v.x; LI[pd(a)] = v.y;
        }
    }
    __syncthreads();

    // ---- apply 16x16 complex group matrices via WMMA ----
    const int ng = passB ? 2 : 3;
    for (int gi = 0; gi < ng; ++gi) {
        const int ls = 4 * (gi + passB);          // log2 element stride in LDS
        const float* Mre = mats + (size_t)(matBase + gi) * 512;
        const float* Mim = Mre + 256;
        const int m  = lane & 15;
        const int hv = lane >> 4;
        // A fragments (16x4 f32: lanes 0-15 K={0,1}, lanes 16-31 K={2,3})
        v2f fr[4], fi[4], fn[4];
        for (int k = 0; k < 4; ++k) {
            int col = 4 * k + 2 * hv;
            fr[k][0] =  Mre[m * 16 + col];  fr[k][1] =  Mre[m * 16 + col + 1];
            fi[k][0] =  Mim[m * 16 + col];  fi[k][1] =  Mim[m * 16 + col + 1];
            fn[k][0] = -fi[k][0];           fn[k][1] = -fi[k][1];
        }
        for (int t = wv; t < 16; t += 8) {         // 2 tiles per wave, uniform
            v8f accR = {0.f,0.f,0.f,0.f,0.f,0.f,0.f,0.f};
            v8f accI = {0.f,0.f,0.f,0.f,0.f,0.f,0.f,0.f};
            const int n   = lane & 15;
            const int vid = t * 16 + n;            // vector (column) id
            const int S   = 1 << ls;
            const int vbase = ((vid >> ls) << (ls + 4)) | (vid & (S - 1));
            for (int k = 0; k < 4; ++k) {
                int a0 = vbase + (4 * k + 2 * hv) * S;
                int a1 = a0 + S;
                v2f xr = { LR[pd(a0)], LR[pd(a1)] };
                v2f xi = { LI[pd(a0)], LI[pd(a1)] };
                accR = __builtin_amdgcn_wmma_f32_16x16x4_f32(false, fr[k], false, xr, (short)0, accR, false, false);
                accR = __builtin_amdgcn_wmma_f32_16x16x4_f32(false, fn[k], false, xi, (short)0, accR, false, false);
                accI = __builtin_amdgcn_wmma_f32_16x16x4_f32(false, fr[k], false, xi, (short)0, accI, false, false);
                accI = __builtin_amdgcn_wmma_f32_16x16x4_f32(false, fi[k], false, xr, (short)0, accI, false, false);
            }
            // D layout: reg v, lane -> D[m = v + 8*(lane>=16)][n = lane&15]
            for (int v = 0; v < 8; ++v) {
                int amp = vbase + (v + 8 * hv) * S;
                LR[pd(amp)] = accR[v];
                LI[pd(amp)] = accI[v];
            }
        }
        __syncthreads();
    }

    // ---- store LDS -> global ----
    if (!passB) {
        const size_t gb = sbase + ((size_t)chunk << 12);
        for (int r = 0; r < 16; ++r) {
            int a = r * 256 + tid;
            float2 v; v.x = LR[pd(a)]; v.y = LI[pd(a)];
            dst[gb + a] = v;
        }
    } else {
        // fused CNOT chain: new[j] = old[j ^ (j>>1)]  ==> scatter old i -> grayDecode(i)
        const int lowBase = chunk << 4;
        for (int r = 0; r < 16; ++r) {
            int a = r * 256 + tid;
            int h = a >> 4, tt = a & 15;
            unsigned x = ((unsigned)h << 12) + (unsigned)lowBase + (unsigned)tt;
            x ^= x >> 1; x ^= x >> 2; x ^= x >> 4; x ^= x >> 8; x ^= x >> 16;
            x &= (DIM - 1);
            float2 v; v.x = LR[pd(a)]; v.y = LI[pd(a)];
            dst[sbase + x] = v;
        }
    }
}

// ---------------------------------------------------------------------------
extern "C" void kernel_launch(void* const* d_in, const int* in_sizes, int n_in,
                              void* d_out, int out_size, void* d_ws, size_t ws_size,
                              hipStream_t stream)
{
    (void)in_sizes; (void)n_in; (void)out_size; (void)ws_size;
    const float* psiRe = (const float*)d_in[0];
    const float* psiIm = (const float*)d_in[1];
    const float* tx    = (const float*)d_in[2];
    const float* tz    = (const float*)d_in[3];

    float*  mats = (float*)d_ws;                       // 20*512 floats = 40 KB
    float2* W    = (float2*)((char*)d_ws + 65536);     // 4*2^20 complex = 33.5 MB
    float2* OUT  = (float2*)d_out;                     // final interleaved complex64

    build_mats<<<20, 256, 0, stream>>>(tx, tz, mats);

    dim3 grid(4 * 256), blk(256);
    // layer 0: A (planes -> W), B (W -> OUT, + CNOT perm)
    qpass<<<grid, blk, 0, stream>>>(psiRe, psiIm, nullptr, W, mats, 0, 0, 1);
    qpass<<<grid, blk, 0, stream>>>(nullptr, nullptr, W, OUT, mats, 3, 1, 0);
    for (int l = 1; l < 4; ++l) {
        qpass<<<grid, blk, 0, stream>>>(nullptr, nullptr, OUT, W, mats, l * 5,     0, 0);
        qpass<<<grid, blk, 0, stream>>>(nullptr, nullptr, W, OUT, mats, l * 5 + 3, 1, 0);
    }
}